// Adj_adjust_88656714924080
// MI455X (gfx1250) — compile-verified
//
#include <hip/hip_runtime.h>
#include <hip/hip_bf16.h>

#define NN 1024
#define DIN 256
#define NSAMP 100

typedef __attribute__((ext_vector_type(2))) float v2f;
typedef __attribute__((ext_vector_type(4))) float v4f;
typedef __attribute__((ext_vector_type(8))) float v8f;

// ---------------------------------------------------------------------------
// Kernel 0: zero the info_loss accumulator slot
// ---------------------------------------------------------------------------
__global__ void zero_scalar_kernel(float* p) {
    if (threadIdx.x == 0) *p = 0.0f;
}

// ---------------------------------------------------------------------------
// Kernel 1: V[i, j] = mean over s of noise[i, s, j]
// noise: [N, S, N] fp32, streamed once (419 MB) -> non-temporal b128 loads.
// One block (256 threads) per row i; each thread owns 4 consecutive j.
// ---------------------------------------------------------------------------
__global__ void noise_mean_kernel(const float* __restrict__ noise,
                                  float* __restrict__ V) {
    const int i  = blockIdx.x;          // row 0..N-1
    const int j4 = threadIdx.x;         // 0..255 -> columns j4*4..j4*4+3
    const v4f* base = (const v4f*)(noise + (size_t)i * NSAMP * NN) + j4;
    v4f acc = {0.f, 0.f, 0.f, 0.f};
    #pragma unroll 4
    for (int s = 0; s < NSAMP; ++s) {
        v4f v = __builtin_nontemporal_load(base + (size_t)s * (NN / 4));
        acc += v;
    }
    const float inv = 1.0f / (float)NSAMP;
    ((v4f*)(V + (size_t)i * NN))[j4] = acc * inv;
}

// ---------------------------------------------------------------------------
// Kernel 2: C = A @ B (fp32 exact) via V_WMMA_F32_16X16X4_F32.
// A: [M,K] row-major, B: [K,N] row-major, C: [M,N] row-major.
// 4 waves per block; each wave owns one 16x16 C tile, tiles share the same
// M-panel of A (L2 reuse). Per K-step of 4:
//   lane l (half = l/16, lm = l%16, kk = kb + 2*half):
//     a = { A[m0+lm][kk], A[m0+lm][kk+1] }   (contiguous v2f load)
//     b = { B[kk][n0+lm], B[kk+1][n0+lm] }
// D/C: VGPR r <-> row m0 + r + 8*half, col n0 + lm.
// ---------------------------------------------------------------------------
__global__ void __launch_bounds__(128)
gemm_f32_wmma_kernel(const float* __restrict__ A,
                     const float* __restrict__ B,
                     float* __restrict__ C,
                     int M, int N, int K) {
    const int lane = threadIdx.x & 31;
    const int wave = threadIdx.x >> 5;
    const int m0 = blockIdx.y * 16;
    const int n0 = (blockIdx.x * 4 + wave) * 16;
    if (m0 >= M || n0 >= N) return;

    const int half = lane >> 4;     // 0 or 1
    const int lm   = lane & 15;

    const float* arow = A + (size_t)(m0 + lm) * K + 2 * half;
    const float* bcol = B + (size_t)(2 * half) * N + (n0 + lm);

    v8f acc = {};
    #pragma unroll 4
    for (int kb = 0; kb < K; kb += 4) {
        v2f a = *(const v2f*)(arow + kb);
        v2f b;
        b.x = bcol[(size_t)kb * N];
        b.y = bcol[(size_t)(kb + 1) * N];
        acc = __builtin_amdgcn_wmma_f32_16x16x4_f32(
            /*neg_a=*/false, a, /*neg_b=*/false, b,
            /*c_mod=*/(short)0, acc, /*reuse_a=*/false, /*reuse_b=*/false);
    }

    float* crow = C + (size_t)(m0 + 8 * half) * N + (n0 + lm);
    #pragma unroll
    for (int r = 0; r < 8; ++r) {
        crow[(size_t)r * N] = acc[r];
    }
}

// ---------------------------------------------------------------------------
// Kernel 3: fused epilogue.
//   p        = sigmoid(h)
//   adj_new  = adj * sigmoid((V + p - 1)/tau)
//   info    += 0.5 * (p - thr)^2      (wave->block->atomic reduction)
// ---------------------------------------------------------------------------
__global__ void __launch_bounds__(256)
epilogue_kernel(const float* __restrict__ h,
                const float* __restrict__ V,
                const float* __restrict__ adj,
                const float* __restrict__ tau_p,
                const float* __restrict__ thr_p,
                float* __restrict__ adj_new,
                float* __restrict__ info,
                int total) {
    const int idx = blockIdx.x * blockDim.x + threadIdx.x;
    const float tau = *tau_p;
    const float thr = *thr_p;
    float local = 0.0f;
    if (idx < total) {
        float hv = h[idx];
        float p  = 1.0f / (1.0f + __expf(-hv));
        float arg = (V[idx] + p - 1.0f) / tau;
        float s  = 1.0f / (1.0f + __expf(-arg));
        adj_new[idx] = adj[idx] * s;
        float d = p - thr;
        local = 0.5f * d * d;
    }
    // wave32 reduction
    #pragma unroll
    for (int off = 16; off > 0; off >>= 1)
        local += __shfl_down(local, off, 32);
    __shared__ float wsum[8];
    const int lane = threadIdx.x & 31;
    const int w    = threadIdx.x >> 5;
    if (lane == 0) wsum[w] = local;
    __syncthreads();
    if (w == 0) {
        float t = (lane < 8) ? wsum[lane] : 0.0f;
        #pragma unroll
        for (int off = 4; off > 0; off >>= 1)
            t += __shfl_down(t, off, 32);
        if (lane == 0) atomicAdd(info, t);
    }
}

// ---------------------------------------------------------------------------
// Launch. d_in: x[N*DIN], W[DIN*N], adj[N*N], noise[N*NSAMP*N], tau, threshold
// d_out: h[N*N] | adj_new[N*N] | info_loss[1]
// d_ws:  V[N*N] | T[N*N]   (8 MB)
// ---------------------------------------------------------------------------
extern "C" void kernel_launch(void* const* d_in, const int* in_sizes, int n_in,
                              void* d_out, int out_size, void* d_ws, size_t ws_size,
                              hipStream_t stream) {
    const float* x     = (const float*)d_in[0];
    const float* W     = (const float*)d_in[1];
    const float* adj   = (const float*)d_in[2];
    const float* noise = (const float*)d_in[3];
    const float* tau_p = (const float*)d_in[4];
    const float* thr_p = (const float*)d_in[5];

    float* h_out   = (float*)d_out;
    float* adj_out = h_out + (size_t)NN * NN;
    float* info    = adj_out + (size_t)NN * NN;

    float* V = (float*)d_ws;                 // [N, N]
    float* T = V + (size_t)NN * NN;          // [N, N]  (x @ W)

    // 0) zero info accumulator
    zero_scalar_kernel<<<1, 32, 0, stream>>>(info);

    // 1) V = mean_s noise[:, s, :]
    noise_mean_kernel<<<NN, NN / 4, 0, stream>>>(noise, V);

    // 2) T = x @ W     (M=1024, N=1024, K=256)
    {
        dim3 grid(NN / 64, NN / 16);
        gemm_f32_wmma_kernel<<<grid, 128, 0, stream>>>(x, W, T, NN, NN, DIN);
    }

    // 3) h = adj @ T   (M=1024, N=1024, K=1024)
    {
        dim3 grid(NN / 64, NN / 16);
        gemm_f32_wmma_kernel<<<grid, 128, 0, stream>>>(adj, T, h_out, NN, NN, NN);
    }

    // 4) fused epilogue
    {
        const int total = NN * NN;
        epilogue_kernel<<<(total + 255) / 256, 256, 0, stream>>>(
            h_out, V, adj, tau_p, thr_p, adj_out, info, total);
    }
}